// GNNClassifier_82154134438100
// MI455X (gfx1250) — compile-verified
//
#include <hip/hip_runtime.h>
#include <hip/hip_bf16.h>

// ---------------------------------------------------------------------------
// GAT GNN classifier for MI455X (gfx1250, wave32).
// Dense GEMMs use V_WMMA_F32_16X16X4_F32 (native fp32 WMMA).
// Both A-tile and transposed weight tile staged in LDS -> inner loop is pure
// ds_load_b64 + v_wmma. Edge softmax / aggregation are L2-resident
// scatter/gather with atomics.
// ---------------------------------------------------------------------------

typedef __attribute__((ext_vector_type(2))) float v2f;
typedef __attribute__((ext_vector_type(8))) float v8f;

#define NEG_SLOPE 0.2f

// ---------------------------------------------------------------------------
// Monotone float<->uint mapping so we can use integer atomicMax for fp max.
// ---------------------------------------------------------------------------
__device__ __forceinline__ unsigned f2mono(float f) {
    unsigned b = __float_as_uint(f);
    return b ^ ((unsigned)(((int)b) >> 31) | 0x80000000u);
}
__device__ __forceinline__ float mono2f(unsigned u) {
    unsigned b = (u & 0x80000000u) ? (u ^ 0x80000000u) : ~u;
    return __uint_as_float(b);
}

// ---------------------------------------------------------------------------
// Fill kernel (bit pattern 0 == mono-key of the -inf sentinel as well)
// ---------------------------------------------------------------------------
__global__ void fill_f32(float* __restrict__ p, float v, long n) {
    long i = (long)blockIdx.x * blockDim.x + threadIdx.x;
    if (i < n) p[i] = v;
}

// ---------------------------------------------------------------------------
// H = X @ W with V_WMMA_F32_16X16X4_F32.
// X: [n, K] row-major, W: [K, NC] row-major, H: [n, NC].
// Block: NC/16 waves; wave w computes the 16x16 tile at cols [16w,16w+16).
//
// LDS staging:
//   ldsA[r][k]  : 16 x K A-tile, padded row stride K+4 (bank-conflict free)
//   ldsBt[c][k] : NC x K transposed weights, padded stride K+2 (8B-aligned
//                 pairs, lanes land 2 banks apart) -> B frags are ds_load_b64
//
// f32 WMMA VGPR layouts (ISA 7.12.2):
//   A 16x4 : lane L, vgpr v -> A[L%16][2*(L/16)+v]
//   B 4x16 : lane L, vgpr v -> B[2*(L/16)+v][L%16]
//   C 16x16: lane L, vgpr v -> C[v+8*(L/16)][L%16]
// ---------------------------------------------------------------------------
template <int K, int NC>
__global__ __launch_bounds__(32 * (NC / 16))
void gat_gemm_wmma(const float* __restrict__ X, const float* __restrict__ W,
                   float* __restrict__ H, int n) {
    constexpr int KPA = K + 4;               // padded A row stride (floats)
    constexpr int KPB = K + 2;               // padded Bt row stride (even!)
    __shared__ float ldsA[16 * KPA];
    __shared__ float ldsBt[NC * KPB];

    const int tid  = threadIdx.x;
    const int wave = tid >> 5;
    const int lane = tid & 31;
    const int half = lane >> 4;
    const int lm   = lane & 15;
    const long row0 = (long)blockIdx.x * 16;

    constexpr int NT = 32 * (NC / 16);

    // Cooperative load of the 16 x K A-tile (coalesced along K).
    for (int i = tid; i < 16 * K; i += NT) {
        int r = i / K, c = i - r * K;
        long node = row0 + r;
        ldsA[r * KPA + c] = (node < n) ? X[node * K + c] : 0.0f;
    }
    // Cooperative transpose of W into LDS (coalesced global reads).
    for (int i = tid; i < K * NC; i += NT) {
        int k = i / NC, c = i - k * NC;
        ldsBt[c * KPB + k] = W[i];
    }
    __syncthreads();

    const int col0 = wave * 16;
    const float* aRow = &ldsA[lm * KPA];
    const float* bRow = &ldsBt[(col0 + lm) * KPB];
    v8f acc = {};

#pragma unroll
    for (int k0 = 0; k0 < K; k0 += 4) {
        const int ka = k0 + 2 * half;
        v2f a, b;
        a.x = aRow[ka];
        a.y = aRow[ka + 1];
        b.x = bRow[ka];
        b.y = bRow[ka + 1];
        // (neg_a, A, neg_b, B, c_mod, C, reuse_a, reuse_b)
        acc = __builtin_amdgcn_wmma_f32_16x16x4_f32(
            false, a, false, b, (short)0, acc, false, false);
    }

#pragma unroll
    for (int v = 0; v < 8; ++v) {
        long node = row0 + v + 8 * half;
        if (node < n) H[node * NC + col0 + lm] = acc[v];
    }
}

// ---------------------------------------------------------------------------
// Per-node attention scores: s_src[i] = h[i]@a_src, s_dst[i] = h[i]@a_dst.
// One wave per node, shuffle reduction (wave32).
// ---------------------------------------------------------------------------
__global__ void node_scores(const float* __restrict__ H,
                            const float* __restrict__ as,
                            const float* __restrict__ ad,
                            float* __restrict__ ssrc, float* __restrict__ sdst,
                            int n, int F) {
    int wid  = (int)((blockIdx.x * (long)blockDim.x + threadIdx.x) >> 5);
    int lane = threadIdx.x & 31;
    if (wid >= n) return;
    float a0 = 0.f, a1 = 0.f;
    const float* h = H + (long)wid * F;
    for (int c = lane; c < F; c += 32) {
        float v = h[c];
        a0 += v * as[c];
        a1 += v * ad[c];
    }
#pragma unroll
    for (int off = 16; off > 0; off >>= 1) {
        a0 += __shfl_down(a0, off, 32);
        a1 += __shfl_down(a1, off, 32);
    }
    if (lane == 0) { ssrc[wid] = a0; sdst[wid] = a1; }
}

// ---------------------------------------------------------------------------
// Edge score + segment max. Edges idx < nE come from the edge list; idx >= nE
// are the PyG-style self-loops (src = dst = idx - nE).
// ---------------------------------------------------------------------------
__global__ void edge_scores_max(const int* __restrict__ src,
                                const int* __restrict__ dst,
                                const float* __restrict__ ssrc,
                                const float* __restrict__ sdst,
                                float* __restrict__ e,
                                unsigned* __restrict__ mkeys,
                                int nE, int nN) {
    int i = blockIdx.x * blockDim.x + threadIdx.x;
    if (i >= nE + nN) return;
    int s, d;
    if (i < nE) { s = src[i]; d = dst[i]; } else { s = d = i - nE; }
    float v = ssrc[s] + sdst[d];
    v = (v > 0.f) ? v : NEG_SLOPE * v;     // leaky_relu
    e[i] = v;
    atomicMax(&mkeys[d], f2mono(v));
}

// ---------------------------------------------------------------------------
// ex = exp(e - max[dst]); den[dst] += ex   (e overwritten with ex)
// ---------------------------------------------------------------------------
__global__ void edge_expsum(const int* __restrict__ dst,
                            float* __restrict__ e,
                            const unsigned* __restrict__ mkeys,
                            float* __restrict__ den, int nE, int nN) {
    int i = blockIdx.x * blockDim.x + threadIdx.x;
    if (i >= nE + nN) return;
    int d = (i < nE) ? dst[i] : i - nE;
    float ex = __expf(e[i] - mono2f(mkeys[d]));
    e[i] = ex;
    atomicAdd(&den[d], ex);
}

// ---------------------------------------------------------------------------
// out[dst] += (ex/den[dst]) * h[src], F channels. One wave per edge; lanes
// cover channels -> coalesced 32-wide atomics (L2-resident).
// ---------------------------------------------------------------------------
template <int F>
__global__ void edge_aggregate(const float* __restrict__ ex,
                               const float* __restrict__ den,
                               const int* __restrict__ src,
                               const int* __restrict__ dst,
                               const float* __restrict__ H,
                               float* __restrict__ OUT, int nE, int nN) {
    int wid  = (int)((blockIdx.x * (long)blockDim.x + threadIdx.x) >> 5);
    int lane = threadIdx.x & 31;
    if (wid >= nE + nN) return;
    int s, d;
    if (wid < nE) { s = src[wid]; d = dst[wid]; } else { s = d = wid - nE; }
    float alpha = ex[wid] / den[d];
    const float* hs = H + (long)s * F;
    float* od = OUT + (long)d * F;
#pragma unroll
    for (int c = lane; c < F; c += 32) atomicAdd(&od[c], alpha * hs[c]);
}

// ---------------------------------------------------------------------------
// out += bias (per channel), optional ReLU
// ---------------------------------------------------------------------------
__global__ void bias_act(float* __restrict__ OUT, const float* __restrict__ bias,
                         long total, int F, int relu) {
    long i = (long)blockIdx.x * blockDim.x + threadIdx.x;
    if (i >= total) return;
    int f = (int)(i % F);
    float v = OUT[i] + bias[f];
    if (relu) v = v > 0.f ? v : 0.f;
    OUT[i] = v;
}

// ---------------------------------------------------------------------------
// Global mean pool: atomics into [G,F] + counts, then divide.
// ---------------------------------------------------------------------------
__global__ void pool_sum(const float* __restrict__ H, const int* __restrict__ batch,
                         float* __restrict__ pooled, float* __restrict__ cnt,
                         int n, int F) {
    long i = (long)blockIdx.x * blockDim.x + threadIdx.x;
    if (i >= (long)n * F) return;
    int node = (int)(i / F), f = (int)(i % F);
    int g = batch[node];
    atomicAdd(&pooled[(long)g * F + f], H[i]);
    if (f == 0) atomicAdd(&cnt[g], 1.0f);
}

__global__ void pool_div(float* __restrict__ pooled, const float* __restrict__ cnt,
                         int nG, int F) {
    int i = blockIdx.x * blockDim.x + threadIdx.x;
    if (i >= nG * F) return;
    float c = cnt[i / F];
    pooled[i] /= (c > 1.f ? c : 1.f);
}

// ---------------------------------------------------------------------------
// MLP head: combined[g] = [e1|e2|e3|scalars] (386) @ W1(386x64) -> relu
//           -> @ W2(64x1) -> sigmoid.  One block (64 threads) per graph.
// ---------------------------------------------------------------------------
__global__ void mlp_head(const float* __restrict__ pooled,   // [3][G][128]
                         const float* __restrict__ scalars,  // [G][2]
                         const float* __restrict__ W1, const float* __restrict__ b1,
                         const float* __restrict__ W2, const float* __restrict__ b2,
                         float* __restrict__ out, int nG) {
    __shared__ float hbuf[64];
    int g = blockIdx.x;
    int j = threadIdx.x;  // 0..63
    float acc = b1[j];
    for (int br = 0; br < 3; ++br) {
        const float* e = pooled + ((long)br * nG + g) * 128;
        for (int i = 0; i < 128; ++i)
            acc += e[i] * W1[(br * 128 + i) * 64 + j];
    }
    acc += scalars[g * 2 + 0] * W1[384 * 64 + j];
    acc += scalars[g * 2 + 1] * W1[385 * 64 + j];
    hbuf[j] = acc > 0.f ? acc : 0.f;
    __syncthreads();
    if (j == 0) {
        float o = b2[0];
        for (int k = 0; k < 64; ++k) o += hbuf[k] * W2[k];
        out[g] = 1.f / (1.f + __expf(-o));
    }
}

// ---------------------------------------------------------------------------
// Host-side orchestration (graph-capture safe: launches only, stream-ordered)
// ---------------------------------------------------------------------------
static inline unsigned gridFor(long total, int block) {
    return (unsigned)((total + block - 1) / block);
}

extern "C" void kernel_launch(void* const* d_in, const int* in_sizes, int n_in,
                              void* d_out, int out_size, void* d_ws, size_t ws_size,
                              hipStream_t stream) {
    const int D_ = 128, HD = 64, OD = 128;
    const int N_ = in_sizes[0] / D_;       // 50000
    const int E_ = in_sizes[1] / 2;        // 800000
    const int G_ = in_sizes[9] / 2;        // 256
    const int nTot = E_ + N_;              // edges + self loops

    // ---- workspace layout (floats) ----
    float*    hbuf   = (float*)d_ws;                         // N*128
    float*    aggbuf = hbuf + (long)N_ * 128;                // N*128
    float*    ssrc   = aggbuf + (long)N_ * 128;              // N
    float*    sdst   = ssrc + N_;                            // N
    unsigned* mkeys  = (unsigned*)(sdst + N_);               // N
    float*    den    = (float*)(mkeys + N_);                 // N
    float*    ebuf   = den + N_;                             // E+N
    float*    pooled = ebuf + nTot;                          // 3*G*128
    float*    cnt    = pooled + 3L * G_ * 128;               // G

    const float* scalars = (const float*)d_in[9];

    for (int br = 0; br < 3; ++br) {
        const float* X     = (const float*)d_in[3 * br + 0];
        const int*   edges = (const int*)d_in[3 * br + 1];   // [2,E] flat
        const int*   batch = (const int*)d_in[3 * br + 2];
        const int*   src = edges;
        const int*   dst = edges + E_;

        const float* W1  = (const float*)d_in[10 + 8 * br + 0];
        const float* as1 = (const float*)d_in[10 + 8 * br + 1];
        const float* ad1 = (const float*)d_in[10 + 8 * br + 2];
        const float* b1  = (const float*)d_in[10 + 8 * br + 3];
        const float* W2  = (const float*)d_in[10 + 8 * br + 4];
        const float* as2 = (const float*)d_in[10 + 8 * br + 5];
        const float* ad2 = (const float*)d_in[10 + 8 * br + 6];
        const float* b2  = (const float*)d_in[10 + 8 * br + 7];

        // ---------------- layer 1: D=128 -> H=64 ----------------
        gat_gemm_wmma<128, 64><<<gridFor(N_, 16), 128, 0, stream>>>(X, W1, hbuf, N_);
        node_scores<<<gridFor((long)N_ * 32, 256), 256, 0, stream>>>(
            hbuf, as1, ad1, ssrc, sdst, N_, HD);
        fill_f32<<<gridFor(N_, 256), 256, 0, stream>>>((float*)mkeys, 0.0f, N_);
        fill_f32<<<gridFor(N_, 256), 256, 0, stream>>>(den, 0.0f, N_);
        edge_scores_max<<<gridFor(nTot, 256), 256, 0, stream>>>(
            src, dst, ssrc, sdst, ebuf, mkeys, E_, N_);
        edge_expsum<<<gridFor(nTot, 256), 256, 0, stream>>>(
            dst, ebuf, mkeys, den, E_, N_);
        fill_f32<<<gridFor((long)N_ * HD, 256), 256, 0, stream>>>(aggbuf, 0.0f, (long)N_ * HD);
        edge_aggregate<64><<<gridFor((long)nTot * 32, 256), 256, 0, stream>>>(
            ebuf, den, src, dst, hbuf, aggbuf, E_, N_);
        bias_act<<<gridFor((long)N_ * HD, 256), 256, 0, stream>>>(
            aggbuf, b1, (long)N_ * HD, HD, 1);

        // ---------------- layer 2: H=64 -> O=128 ----------------
        gat_gemm_wmma<64, 128><<<gridFor(N_, 16), 256, 0, stream>>>(aggbuf, W2, hbuf, N_);
        node_scores<<<gridFor((long)N_ * 32, 256), 256, 0, stream>>>(
            hbuf, as2, ad2, ssrc, sdst, N_, OD);
        fill_f32<<<gridFor(N_, 256), 256, 0, stream>>>((float*)mkeys, 0.0f, N_);
        fill_f32<<<gridFor(N_, 256), 256, 0, stream>>>(den, 0.0f, N_);
        edge_scores_max<<<gridFor(nTot, 256), 256, 0, stream>>>(
            src, dst, ssrc, sdst, ebuf, mkeys, E_, N_);
        edge_expsum<<<gridFor(nTot, 256), 256, 0, stream>>>(
            dst, ebuf, mkeys, den, E_, N_);
        fill_f32<<<gridFor((long)N_ * OD, 256), 256, 0, stream>>>(aggbuf, 0.0f, (long)N_ * OD);
        edge_aggregate<128><<<gridFor((long)nTot * 32, 256), 256, 0, stream>>>(
            ebuf, den, src, dst, hbuf, aggbuf, E_, N_);
        bias_act<<<gridFor((long)N_ * OD, 256), 256, 0, stream>>>(
            aggbuf, b2, (long)N_ * OD, OD, 0);

        // ---------------- mean pool ----------------
        float* pooled_b = pooled + (long)br * G_ * 128;
        fill_f32<<<gridFor((long)G_ * 128, 256), 256, 0, stream>>>(pooled_b, 0.0f, (long)G_ * 128);
        fill_f32<<<gridFor(G_, 256), 256, 0, stream>>>(cnt, 0.0f, G_);
        pool_sum<<<gridFor((long)N_ * 128, 256), 256, 0, stream>>>(
            aggbuf, batch, pooled_b, cnt, N_, 128);
        pool_div<<<gridFor((long)G_ * 128, 256), 256, 0, stream>>>(pooled_b, cnt, G_, 128);
    }

    // ---------------- MLP head ----------------
    const float* mlpW1 = (const float*)d_in[34];
    const float* mlpb1 = (const float*)d_in[35];
    const float* mlpW2 = (const float*)d_in[36];
    const float* mlpb2 = (const float*)d_in[37];
    mlp_head<<<G_, 64, 0, stream>>>(pooled, scalars, mlpW1, mlpb1, mlpW2, mlpb2,
                                    (float*)d_out, G_);
}